// CRFBiRNNModel_41429254537332
// MI455X (gfx1250) — compile-verified
//
#include <hip/hip_runtime.h>

typedef __attribute__((ext_vector_type(2))) float v2f;
typedef __attribute__((ext_vector_type(8))) float v8f;

#define NSEQ 512
#define NI   513
#define KT   12
#define DD   256
#define HH   16
#define NEGV (-1000000000.0f)

// workspace layout (float offsets)
#define OFF_WF     0         // 528*256 padded Wf
#define OFF_BCAT   135168    // 256*48 packed [Mw.T | MPw.T | UBw.T | pad]
#define OFF_WPROJ  147456    // 528*48
#define OFF_WBC    172800    // 512*12
#define OFF_SA     178944    // 12*12
#define OFF_TA     179088
#define OFF_TB     179232
#define OFF_HS     179376    // 513*16
#define OFF_HP     187584    // 513*16
#define OFF_BASEA  195792    // 6156
#define OFF_BASEB  201948    // 6156
#define OFF_EMIT   208104    // 6144
#define OFF_LOGPHI 214248    // 6156*12
// total 288120 floats (~1.15 MB)

__device__ __forceinline__ float sigf(float x) {
    return 1.0f / (1.0f + __expf(-x));
}

__device__ __forceinline__ v8f wmma4(v2f a, v2f b, v8f c) {
    // V_WMMA_F32_16X16X4_F32 : D = A(16x4,f32) * B(4x16,f32) + C(16x16,f32)
    return __builtin_amdgcn_wmma_f32_16x16x4_f32(false, a, false, b, (short)0, c,
                                                 false, false);
}

// ---- gather Wf = [E[words]; 0] padded to 528 rows -------------------------
__global__ void k_gather(const float* __restrict__ E, const int* __restrict__ words,
                         float* __restrict__ ws) {
    int idx = blockIdx.x * blockDim.x + threadIdx.x;
    if (idx >= 528 * DD) return;
    int row = idx / DD, col = idx % DD;
    float v = 0.0f;
    if (row < NSEQ) v = E[(long)words[row] * DD + col];
    ws[OFF_WF + idx] = v;
}

// ---- pack Bcat(256x48): cols 0-15 Mw.T, 16-31 MPw.T, 32-43 UBw.T ----------
__global__ void k_packB(const float* __restrict__ M, const float* __restrict__ MP,
                        const float* __restrict__ UB, float* __restrict__ ws) {
    int idx = blockIdx.x * blockDim.x + threadIdx.x;
    if (idx >= DD * 48) return;
    int k = idx / 48, n = idx % 48;
    float v = 0.0f;
    if (n < 16)      v = M[n * 273 + 17 + k];          // Mw[n][k]
    else if (n < 32) v = MP[(n - 16) * 273 + 1 + k];   // MPw[n'][k]
    else if (n < 44) v = UB[(n - 32) * 545 + 273 + k]; // UBw[a][k]
    ws[OFF_BCAT + idx] = v;
}

// ---- WBc[words[i]][t] = sum_s W_PhiB[words[i]].reshape(12,12)[s][t] -------
__global__ void k_wbc(const float* __restrict__ WPhiB, const int* __restrict__ words,
                      float* __restrict__ ws) {
    int idx = blockIdx.x * blockDim.x + threadIdx.x;
    if (idx >= NSEQ * KT) return;
    int i = idx / KT, t = idx % KT;
    const float* row = WPhiB + (long)words[i] * 144;
    float s = 0.0f;
#pragma unroll
    for (int q = 0; q < KT; ++q) s += row[q * KT + t];
    ws[OFF_WBC + idx] = s;
}

// ---- WMMA: Wproj(528x48) = Wf(528x256) @ Bcat(256x48) ---------------------
__global__ void k_proj(float* __restrict__ ws) {
    int lane  = threadIdx.x;
    int mtile = blockIdx.x;            // 0..32
    int ntile = blockIdx.y;            // 0..2
    int mrow  = lane & 15;
    int khalf = (lane >> 4) << 1;      // 0 or 2
    const float* A = ws + OFF_WF + (mtile * 16 + mrow) * DD;
    const float* B = ws + OFF_BCAT;
    int nc = ntile * 16 + (lane & 15);
    v8f c = (v8f)0.0f;
    for (int k0 = 0; k0 < DD; k0 += 4) {
        int ka = k0 + khalf;
        v2f av, bv;
        av.x = A[ka];
        av.y = A[ka + 1];
        bv.x = B[ka * 48 + nc];
        bv.y = B[(ka + 1) * 48 + nc];
        c = wmma4(av, bv, c);
    }
    float* Wp = ws + OFF_WPROJ;
#pragma unroll
    for (int v = 0; v < 8; ++v) {
        int row = mtile * 16 + v + ((lane >= 16) ? 8 : 0);
        Wp[row * 48 + nc] = c[v];      // padded buffer: always in bounds
    }
}

// ---- WMMA: SA=UAs@T.T, TA=UAt@T.T, TB=UBt@T.T (12x12 each, K=256) ---------
__global__ void k_small(const float* __restrict__ UA, const float* __restrict__ UB,
                        const float* __restrict__ T, float* __restrict__ ws) {
    int lane  = threadIdx.x;
    int which = blockIdx.x;            // 0:SA 1:TA 2:TB
    const float* src = (which == 2) ? UB : UA;
    int coff = (which == 1) ? 273 : 17;
    float* dst = ws + ((which == 0) ? OFF_SA : (which == 1) ? OFF_TA : OFF_TB);
    int mrow  = lane & 15;
    int khalf = (lane >> 4) << 1;
    int nc    = lane & 15;
    v8f c = (v8f)0.0f;
    for (int k0 = 0; k0 < DD; k0 += 4) {
        int ka = k0 + khalf;
        v2f av, bv;
        av.x = (mrow < KT) ? src[mrow * 545 + coff + ka]     : 0.0f;
        av.y = (mrow < KT) ? src[mrow * 545 + coff + ka + 1] : 0.0f;
        bv.x = (nc < KT)   ? T[nc * DD + ka]                 : 0.0f;
        bv.y = (nc < KT)   ? T[nc * DD + ka + 1]             : 0.0f;
        c = wmma4(av, bv, c);
    }
#pragma unroll
    for (int v = 0; v < 8; ++v) {
        int row = v + ((lane >= 16) ? 8 : 0);
        if (row < KT && nc < KT) dst[row * KT + nc] = c[v];
    }
}

// ---- sequential RNN scans: wave0 forward (hs), wave1 backward (hp) --------
__global__ void k_scans(const float* __restrict__ M, const float* __restrict__ MP,
                        float* __restrict__ ws) {
    int tid  = threadIdx.x;
    int wave = tid >> 5;
    int lane = tid & 31;
    const float* Wproj = ws + OFF_WPROJ;
    if (wave == 0) {
        if (lane < HH) {
            float mh[HH];
#pragma unroll
            for (int m = 0; m < HH; ++m) mh[m] = M[lane * 273 + 1 + m];
            float m0l = M[lane * 273];
            float h = 0.0f;
            float* hs = ws + OFF_HS;
            for (int j = 0; j < NI; ++j) {
                float pf = Wproj[j * 48 + lane] + m0l;
                float dot = 0.0f;
#pragma unroll
                for (int m = 0; m < HH; ++m) dot += mh[m] * __shfl(h, m, 32);
                h = sigf(pf + dot);
                hs[j * HH + lane] = h;
            }
        }
    } else {
        if (lane < HH) {
            float mph[HH];
#pragma unroll
            for (int m = 0; m < HH; ++m) mph[m] = MP[lane * 273 + 257 + m];
            float mp0l = MP[lane * 273];
            float hn = sigf(mp0l);
            float* hp = ws + OFF_HP;
            hp[NSEQ * HH + lane] = hn;   // hp[512] = hp_n
            hp[lane] = 0.0f;             // hp[0]   = 0
            float h = hn;
            for (int j = NSEQ - 2; j >= 0; --j) {       // j = 510..0
                float pb = Wproj[(j + 1) * 48 + 16 + lane] + mp0l;
                float dot = 0.0f;
#pragma unroll
                for (int m = 0; m < HH; ++m) dot += mph[m] * __shfl(h, m, 32);
                h = sigf(pb + dot);
                hp[(j + 1) * HH + lane] = h;            // hps[j] -> hp[j+1]
            }
        }
    }
}

// ---- baseA/baseB per (i,a) ------------------------------------------------
__global__ void k_base(const float* __restrict__ UA, const float* __restrict__ UB,
                       float* __restrict__ ws) {
    int r = blockIdx.x * blockDim.x + threadIdx.x;
    if (r >= NI * KT) return;
    int i = r / KT, a = r % KT;
    const float* hs = ws + OFF_HS + i * HH;
    const float* hp = ws + OFF_HP;
    const float* ua = UA + a * 545;
    const float* ub = UB + a * 545;
    float bA = ua[0], bB = ub[0];
#pragma unroll
    for (int h = 0; h < HH; ++h) {
        float hv = hs[h];
        bA += hv * ua[1 + h];
        bB += hv * ub[1 + h];
    }
    if (i >= 2) {
        const float* hpa = hp + (i - 2) * HH;   // hpA[i] = hp[i-2]
#pragma unroll
        for (int h = 0; h < HH; ++h) bA += hpa[h] * ua[529 + h];
    }
    if (i >= 1) {
        const float* hpb = hp + (i - 1) * HH;   // hpB[i] = hp[i-1]
#pragma unroll
        for (int h = 0; h < HH; ++h) bB += hpb[h] * ub[529 + h];
    }
    bB += ws[OFF_WPROJ + i * 48 + 32 + a];      // Wf @ UBw.T column
    ws[OFF_BASEA + r] = bA;
    ws[OFF_BASEB + r] = bB;
}

// ---- emit[i,a] = sum_t sigmoid(baseB[i,a]+TB[a,t]) * WBc[i,t] -------------
__global__ void k_emit(float* __restrict__ ws) {
    int r = blockIdx.x * blockDim.x + threadIdx.x;
    if (r >= NSEQ * KT) return;
    int i = r / KT, a = r % KT;
    float bB = ws[OFF_BASEB + r];
    const float* tb  = ws + OFF_TB + a * KT;
    const float* wbc = ws + OFF_WBC + i * KT;
    float s = 0.0f;
#pragma unroll
    for (int t = 0; t < KT; ++t) s += sigf(bB + tb[t]) * wbc[t];
    ws[OFF_EMIT + r] = s;
}

// ---- WMMA with fused sigmoid A-fragment:
//      logphi(6156x12) = fA(6156x144) @ W_PhiA.T(144x12)
//      fA[r=(i,a)][c=(s,t)] = sigmoid(baseA[r] + SA[a,s] + TA[a,t]) ---------
__global__ void k_phi(const float* __restrict__ WPhiA, float* __restrict__ ws) {
    int lane  = threadIdx.x;
    int mtile = blockIdx.x;            // 0..384
    int mrow  = lane & 15;
    int khalf = (lane >> 4) << 1;
    int r = mtile * 16 + mrow;
    bool valid = (r < NI * KT);
    int a = r % KT;
    const float* baseA = ws + OFF_BASEA;
    const float* SA = ws + OFF_SA + a * KT;
    const float* TA = ws + OFF_TA + a * KT;
    float bA = valid ? baseA[r] : 0.0f;
    int nc = lane & 15;
    v8f c = (v8f)0.0f;
    for (int k0 = 0; k0 < 144; k0 += 4) {
        int c0 = k0 + khalf;
        int c1 = c0 + 1;
        v2f av, bv;
        av.x = valid ? sigf(bA + SA[c0 / 12] + TA[c0 % 12]) : 0.0f;
        av.y = valid ? sigf(bA + SA[c1 / 12] + TA[c1 % 12]) : 0.0f;
        bv.x = (nc < KT) ? WPhiA[nc * 144 + c0] : 0.0f;
        bv.y = (nc < KT) ? WPhiA[nc * 144 + c1] : 0.0f;
        c = wmma4(av, bv, c);
    }
    float* logphi = ws + OFF_LOGPHI;
#pragma unroll
    for (int v = 0; v < 8; ++v) {
        int row = mtile * 16 + v + ((lane >= 16) ? 8 : 0);
        if (row < NI * KT && nc < KT) logphi[row * KT + nc] = c[v];
    }
}

// ---- sequential CRF forward recursion + final logsumexp (single wave) -----
__global__ void k_crf(const int* __restrict__ tags, const int* __restrict__ eosp,
                      const float* __restrict__ ws, float* __restrict__ out) {
    int lane = threadIdx.x;
    int eos = eosp[0];
    const float* logphi = ws + OFF_LOGPHI;
    const float* emit   = ws + OFF_EMIT;
    float a_ = (lane == eos) ? 0.0f : NEGV;
    float az = a_;
    for (int i = 0; i < NSEQ; ++i) {
        int tg = tags[i];
        float em = (lane < KT) ? emit[i * KT + lane] : 0.0f;
        float vz[KT], va[KT];
        float mz = -3.0e38f, ma = -3.0e38f;
#pragma unroll
        for (int s = 0; s < KT; ++s) {
            float azs = __shfl(az, s, 32);
            float as  = __shfl(a_, s, 32);
            float ph  = (lane < KT) ? logphi[(i * KT + s) * KT + lane] : NEGV;
            float v1 = azs + ph; vz[s] = v1; mz = fmaxf(mz, v1);
            float v2 = as  + ph; va[s] = v2; ma = fmaxf(ma, v2);
        }
        float sz = 0.0f, sa = 0.0f;
#pragma unroll
        for (int s = 0; s < KT; ++s) {
            sz += __expf(vz[s] - mz);
            sa += __expf(va[s] - ma);
        }
        float naz = __logf(sz) + mz + em;
        float na  = __logf(sa) + ma + em;
        az = naz;
        a_ = (lane == tg) ? na : NEGV;
    }
    // last[t] = logphiA[512, t, eos]
    float last = (lane < KT) ? logphi[(NSEQ * KT + lane) * KT + eos] : NEGV;
    float x1 = a_ + last, x2 = az + last;
    float m1 = -3.0e38f, m2 = -3.0e38f;
#pragma unroll
    for (int s = 0; s < KT; ++s) {
        m1 = fmaxf(m1, __shfl(x1, s, 32));
        m2 = fmaxf(m2, __shfl(x2, s, 32));
    }
    float s1 = 0.0f, s2 = 0.0f;
#pragma unroll
    for (int s = 0; s < KT; ++s) {
        s1 += __expf(__shfl(x1, s, 32) - m1);
        s2 += __expf(__shfl(x2, s, 32) - m2);
    }
    if (lane == 0) out[0] = (__logf(s1) + m1) - (__logf(s2) + m2);
}

extern "C" void kernel_launch(void* const* d_in, const int* in_sizes, int n_in,
                              void* d_out, int out_size, void* d_ws, size_t ws_size,
                              hipStream_t stream) {
    const float* E     = (const float*)d_in[0];
    const float* M     = (const float*)d_in[1];
    const float* MP    = (const float*)d_in[2];
    const float* T     = (const float*)d_in[3];
    const float* UA    = (const float*)d_in[4];
    const float* UB    = (const float*)d_in[5];
    const float* WPhiA = (const float*)d_in[6];
    const float* WPhiB = (const float*)d_in[7];
    const int*   words = (const int*)d_in[8];
    const int*   tags  = (const int*)d_in[9];
    const int*   eos   = (const int*)d_in[10];
    float* out = (float*)d_out;
    float* ws  = (float*)d_ws;

    (void)in_sizes; (void)n_in; (void)out_size; (void)ws_size;

    k_gather<<<(528 * DD + 255) / 256, 256, 0, stream>>>(E, words, ws);
    k_packB<<<(DD * 48 + 255) / 256, 256, 0, stream>>>(M, MP, UB, ws);
    k_wbc<<<(NSEQ * KT + 255) / 256, 256, 0, stream>>>(WPhiB, words, ws);

    k_proj<<<dim3(33, 3), 32, 0, stream>>>(ws);
    k_small<<<3, 32, 0, stream>>>(UA, UB, T, ws);

    k_scans<<<1, 64, 0, stream>>>(M, MP, ws);

    k_base<<<(NI * KT + 255) / 256, 256, 0, stream>>>(UA, UB, ws);
    k_emit<<<(NSEQ * KT + 255) / 256, 256, 0, stream>>>(ws);

    k_phi<<<385, 32, 0, stream>>>(WPhiA, ws);

    k_crf<<<1, 32, 0, stream>>>(tags, eos, ws, out);
}